// HGAT_17119739642018
// MI455X (gfx1250) — compile-verified
//
#include <hip/hip_runtime.h>
#include <hip/hip_bf16.h>
#include <math.h>

// ---------------------------------------------------------------------------
// HGAT (2-layer) for MI455X / gfx1250.
// N=100000 nodes, M=20000 hyperedges, E=1.6M edges, D=128.
// Memory-bound on the 4 edge gather/scatter passes (~3.3 TB @ 23.3 TB/s).
// GEMMs use V_WMMA_F32_16X16X4_F32 (full fp32 precision). ReLU is a template
// parameter so layer 1's GEMM inner loop is pure load->wmma (no cndmask).
// ---------------------------------------------------------------------------

#define HGAT_N 100000
#define HGAT_M 20000
#define HGAT_E 1600000
#define HGAT_D 128

typedef float v2f __attribute__((ext_vector_type(2)));
typedef float v8f __attribute__((ext_vector_type(8)));

// --- monotone float<->uint mapping so atomicMax(uint) orders like floats ----
__device__ __forceinline__ unsigned enc_f32(float f) {
    unsigned u = __float_as_uint(f);
    return (u & 0x80000000u) ? ~u : (u | 0x80000000u);
}
__device__ __forceinline__ float dec_f32(unsigned u) {
    return __uint_as_float((u & 0x80000000u) ? (u ^ 0x80000000u) : ~u);
}

// ---------------------------------------------------------------------------
// GEMM: H[r, c] = sum_k relu?(X[r,k]) * W[k,c] + b[c], D=128, WMMA f32 16x16x4.
// One wave per 16x16 output tile; 8 waves (256 threads) per block.
// A layout: lane%16 = M row, lane/16 selects K pair -> v2f per lane.
// B layout: lane%16 = N col, lane/16 selects K pair -> v2f per lane.
// C/D layout: vgpr v -> M = v + 8*(lane/16), N = lane%16.
// ---------------------------------------------------------------------------
template <int RELU>
__global__ __launch_bounds__(256)
void hgat_gemm_wmma(const float* __restrict__ X, const float* __restrict__ W,
                    const float* __restrict__ bias, float* __restrict__ H,
                    int nRows)
{
    const int lane = threadIdx.x & 31;
    const int wave = threadIdx.x >> 5;
    const int tile = blockIdx.x * 8 + wave;
    const int colTiles = HGAT_D / 16;                // 8
    const int rowTile = tile / colTiles;
    const int colTile = tile % colTiles;
    if (rowTile >= nRows / 16) return;               // wave-uniform exit

    const int half = lane >> 4;                      // 0: K pair {0,1}, 1: {2,3}
    const int l16  = lane & 15;
    const int rowA = rowTile * 16 + l16;             // A-matrix row for this lane
    const int colB = colTile * 16 + l16;             // B-matrix col for this lane

    const float* __restrict__ xrow = X + (size_t)rowA * HGAT_D;

    v8f acc = {0.f, 0.f, 0.f, 0.f, 0.f, 0.f, 0.f, 0.f};
    #pragma unroll 4
    for (int k = 0; k < HGAT_D; k += 4) {
        const int k0 = k + 2 * half;
        float a0 = xrow[k0];
        float a1 = xrow[k0 + 1];
        if (RELU) { a0 = fmaxf(a0, 0.f); a1 = fmaxf(a1, 0.f); }
        v2f a; a.x = a0; a.y = a1;
        v2f b;
        b.x = W[(size_t)k0 * HGAT_D + colB];
        b.y = W[(size_t)(k0 + 1) * HGAT_D + colB];
        // D = A(16x4) * B(4x16) + C  -> v_wmma_f32_16x16x4_f32
        acc = __builtin_amdgcn_wmma_f32_16x16x4_f32(
            /*neg_a=*/false, a, /*neg_b=*/false, b,
            /*c_mod=*/(short)0, acc, /*reuse_a=*/false, /*reuse_b=*/false);
    }

    const float bc = bias[colB];
    const int rBase = rowTile * 16 + half * 8;       // C layout: M = v + 8*half
    #pragma unroll
    for (int v = 0; v < 8; ++v) {
        H[(size_t)(rBase + v) * HGAT_D + colB] = acc[v] + bc;
    }
}

// ---------------------------------------------------------------------------
// Row dot product: out[r] = dot(H[r, :], vec), one wave per row.
// ---------------------------------------------------------------------------
__global__ __launch_bounds__(256)
void hgat_rowdot(const float* __restrict__ H, const float* __restrict__ vec,
                 float* __restrict__ out, int nRows)
{
    const int lane = threadIdx.x & 31;
    const int row  = (int)((blockIdx.x * (unsigned)blockDim.x + threadIdx.x) >> 5);
    if (row >= nRows) return;
    const float* __restrict__ hr = H + (size_t)row * HGAT_D;
    float s = 0.f;
    #pragma unroll
    for (int j = 0; j < 4; ++j)
        s += hr[lane + 32 * j] * vec[lane + 32 * j];
    #pragma unroll
    for (int off = 16; off > 0; off >>= 1)
        s += __shfl_xor(s, off, 32);
    if (lane == 0) out[row] = s;
}

// ---------------------------------------------------------------------------
// Global-softmax pass 1: max over edge scores e_k = s_src[src[k]] (+ s_dst[bidx[k]]).
// ---------------------------------------------------------------------------
__global__ __launch_bounds__(256)
void hgat_edge_max(const int* __restrict__ src, const int* __restrict__ bidx,
                   const float* __restrict__ s_src, const float* __restrict__ s_dst,
                   unsigned* __restrict__ maxbits, int nE)
{
    __shared__ unsigned sm[256];
    unsigned best = 0u;                              // encodes below -inf
    const int stride = gridDim.x * blockDim.x;
    for (int k = blockIdx.x * blockDim.x + threadIdx.x; k < nE; k += stride) {
        float e = s_src[src[k]];
        if (s_dst) e += s_dst[bidx[k]];
        const unsigned u = enc_f32(e);
        best = best > u ? best : u;
    }
    sm[threadIdx.x] = best;
    __syncthreads();
    for (int s = 128; s > 0; s >>= 1) {
        if ((int)threadIdx.x < s) {
            const unsigned o = sm[threadIdx.x + s];
            if (o > sm[threadIdx.x]) sm[threadIdx.x] = o;
        }
        __syncthreads();
    }
    if (threadIdx.x == 0) atomicMax(maxbits, sm[0]);
}

// ---------------------------------------------------------------------------
// Global-softmax pass 2: sum of exp(e_k - max).
// ---------------------------------------------------------------------------
__global__ __launch_bounds__(256)
void hgat_edge_sum(const int* __restrict__ src, const int* __restrict__ bidx,
                   const float* __restrict__ s_src, const float* __restrict__ s_dst,
                   const unsigned* __restrict__ maxbits, float* __restrict__ sum,
                   int nE)
{
    __shared__ float sm[256];
    const float m = dec_f32(*maxbits);
    float acc = 0.f;
    const int stride = gridDim.x * blockDim.x;
    for (int k = blockIdx.x * blockDim.x + threadIdx.x; k < nE; k += stride) {
        float e = s_src[src[k]];
        if (s_dst) e += s_dst[bidx[k]];
        acc += __expf(e - m);
    }
    sm[threadIdx.x] = acc;
    __syncthreads();
    for (int s = 128; s > 0; s >>= 1) {
        if ((int)threadIdx.x < s) sm[threadIdx.x] += sm[threadIdx.x + s];
        __syncthreads();
    }
    if (threadIdx.x == 0) atomicAdd(sum, sm[0]);
}

// ---------------------------------------------------------------------------
// Edge scatter: featDst[dst[k], :] += softmax_k * featSrc[src[k], :].
// One wave per edge; lane handles 4 contiguous columns (float4 gather,
// 4x global_atomic_add_f32 scatter).
// ---------------------------------------------------------------------------
__global__ __launch_bounds__(256)
void hgat_edge_scatter(const int* __restrict__ src, const int* __restrict__ dst,
                       const int* __restrict__ bidx,
                       const float* __restrict__ s_src, const float* __restrict__ s_dst,
                       const float* __restrict__ featSrc, float* __restrict__ featDst,
                       const unsigned* __restrict__ maxbits,
                       const float* __restrict__ sum, int nE)
{
    const int lane = threadIdx.x & 31;
    const long long flat = (long long)blockIdx.x * blockDim.x + threadIdx.x;
    const int edge = (int)(flat >> 5);
    if (edge >= nE) return;

    const int si = src[edge];
    const int di = dst[edge];
    float e = s_src[si];
    if (s_dst) e += s_dst[bidx[edge]];
    const float m = dec_f32(*maxbits);
    const float w = __expf(e - m) / (*sum);

    const float4 hv = *(const float4*)(featSrc + (size_t)si * HGAT_D + lane * 4);
    float* __restrict__ o = featDst + (size_t)di * HGAT_D + lane * 4;
    atomicAdd(o + 0, w * hv.x);
    atomicAdd(o + 1, w * hv.y);
    atomicAdd(o + 2, w * hv.z);
    atomicAdd(o + 3, w * hv.w);
}

// ---------------------------------------------------------------------------
// Host-side orchestration (all launches on `stream`; graph-capture safe).
// ---------------------------------------------------------------------------
extern "C" void kernel_launch(void* const* d_in, const int* in_sizes, int n_in,
                              void* d_out, int out_size, void* d_ws, size_t ws_size,
                              hipStream_t stream) {
    (void)in_sizes; (void)n_in; (void)out_size;

    const float* x       = (const float*)d_in[0];
    const int*   in_src  = (const int*)  d_in[1];
    const int*   in_dst  = (const int*)  d_in[2];
    const int*   has_src = (const int*)  d_in[3];
    const int*   has_dst = (const int*)  d_in[4];
    const float* W1      = (const float*)d_in[5];
    const float* b1      = (const float*)d_in[6];
    const float* an1     = (const float*)d_in[7];
    const float* ae1     = (const float*)d_in[8];
    const float* W2      = (const float*)d_in[9];
    const float* b2      = (const float*)d_in[10];
    const float* an2     = (const float*)d_in[11];
    const float* ae2     = (const float*)d_in[12];
    float* out = (float*)d_out;

    // workspace layout (floats)
    const size_t szH  = (size_t)HGAT_N * HGAT_D;     // 12.8M
    const size_t szHE = (size_t)HGAT_M * HGAT_D;     // 2.56M
    float* h      = (float*)d_ws;                    // [N,D] per-layer transform
    float* hn1    = h      + szH;                    // [N,D] layer-1 output
    float* he     = hn1    + szH;                    // [M,D] hyperedge features
    float* n_attn = he     + szHE;                   // [N]
    float* e_attn = n_attn + HGAT_N;                 // [M]
    unsigned* scal = (unsigned*)(e_attn + HGAT_M);   // 8 slots: {max,sum} x4
    const size_t needBytes = ((size_t)(2 * szH + szHE) + HGAT_N + HGAT_M + 8) * 4;
    if (ws_size < needBytes) return;                 // safety; ws expected larger

    // grids
    const int gemmBlocks    = ((HGAT_N / 16) * (HGAT_D / 16)) / 8;   // 6250
    const int rowdotBlocksN = (HGAT_N + 7) / 8;
    const int rowdotBlocksM = (HGAT_M + 7) / 8;
    const int redBlocks     = 2048;
    const int scatBlocks    = (HGAT_E + 7) / 8;                      // 1 wave/edge

    // per-call initialization (scatter targets + reduction scalars)
    hipMemsetAsync(hn1,  0, szH  * 4, stream);
    hipMemsetAsync(out,  0, szH  * 4, stream);
    hipMemsetAsync(scal, 0, 8 * 4,    stream);

    for (int layer = 0; layer < 2; ++layer) {
        const float* Xin = (layer == 0) ? x   : hn1;
        const float* Wm  = (layer == 0) ? W1  : W2;
        const float* bv  = (layer == 0) ? b1  : b2;
        const float* an  = (layer == 0) ? an1 : an2;
        const float* ae  = (layer == 0) ? ae1 : ae2;
        float* hn        = (layer == 0) ? hn1 : out;
        unsigned* mx1 = scal + 4 * layer + 0;
        float*    sm1 = (float*)(scal + 4 * layer + 1);
        unsigned* mx2 = scal + 4 * layer + 2;
        float*    sm2 = (float*)(scal + 4 * layer + 3);

        // h = (relu?)Xin @ W + b            (WMMA f32; ReLU fused only in layer 2)
        if (layer == 0)
            hgat_gemm_wmma<0><<<gemmBlocks, 256, 0, stream>>>(Xin, Wm, bv, h, HGAT_N);
        else
            hgat_gemm_wmma<1><<<gemmBlocks, 256, 0, stream>>>(Xin, Wm, bv, h, HGAT_N);

        // n_attn = h @ an
        hgat_rowdot<<<rowdotBlocksN, 256, 0, stream>>>(h, an, n_attn, HGAT_N);

        // softmax over e = n_attn[in_src]
        hgat_edge_max<<<redBlocks, 256, 0, stream>>>(in_src, nullptr, n_attn,
                                                     nullptr, mx1, HGAT_E);
        hgat_edge_sum<<<redBlocks, 256, 0, stream>>>(in_src, nullptr, n_attn,
                                                     nullptr, mx1, sm1, HGAT_E);
        // h_e = segment_sum(attn * h[in_src], in_dst)
        hipMemsetAsync(he, 0, szHE * 4, stream);
        hgat_edge_scatter<<<scatBlocks, 256, 0, stream>>>(in_src, in_dst, nullptr,
                                                          n_attn, nullptr, h, he,
                                                          mx1, sm1, HGAT_E);
        // e_attn = h_e @ ae
        hgat_rowdot<<<rowdotBlocksM, 256, 0, stream>>>(he, ae, e_attn, HGAT_M);

        // softmax over e2 = e_attn[has_src] + n_attn[has_dst]
        hgat_edge_max<<<redBlocks, 256, 0, stream>>>(has_src, has_dst, e_attn,
                                                     n_attn, mx2, HGAT_E);
        hgat_edge_sum<<<redBlocks, 256, 0, stream>>>(has_src, has_dst, e_attn,
                                                     n_attn, mx2, sm2, HGAT_E);
        // h_n = segment_sum(attn2 * h_e[has_src], has_dst)
        hgat_edge_scatter<<<scatBlocks, 256, 0, stream>>>(has_src, has_dst, has_dst,
                                                          e_attn, n_attn, he, hn,
                                                          mx2, sm2, HGAT_E);
        // layer-1 ReLU is fused into layer-2 GEMM input read
    }
}